// Sector4Pooling2D_24867860644147
// MI455X (gfx1250) — compile-verified
//
#include <hip/hip_runtime.h>
#include <stdint.h>

#ifndef __has_builtin
#define __has_builtin(x) 0
#endif

#if __has_builtin(__builtin_amdgcn_global_load_async_to_lds_b128) && \
    __has_builtin(__builtin_amdgcn_global_store_async_from_lds_b128)
#define USE_ASYNC_BUILTIN 1
#else
#define USE_ASYNC_BUILTIN 0
#endif

typedef int v4i __attribute__((vector_size(16)));
typedef __attribute__((address_space(1))) v4i as1_v4i;   // "__device__ int4*" in clang diagnostics
typedef __attribute__((address_space(3))) v4i as3_v4i;   // "__shared__ int4*" in clang diagnostics
typedef __attribute__((address_space(3))) char as3_char;

// 16-byte async DMA: global -> LDS (no VGPR data round-trip, tracked by ASYNCcnt)
__device__ __forceinline__ void async_g2l_b128(const void* gsrc, void* ldst) {
#if USE_ASYNC_BUILTIN
  __builtin_amdgcn_global_load_async_to_lds_b128((as1_v4i*)gsrc, (as3_v4i*)ldst, 0, 0);
#else
  unsigned loff = (unsigned)(uintptr_t)(as3_char*)ldst;
  asm volatile("global_load_async_to_lds_b128 %0, %1, off"
               :: "v"(loff), "v"(gsrc)
               : "memory");
#endif
}

// 16-byte async DMA: LDS -> global
__device__ __forceinline__ void async_l2g_b128(void* gdst, const void* lsrc) {
#if USE_ASYNC_BUILTIN
  __builtin_amdgcn_global_store_async_from_lds_b128((as1_v4i*)gdst, (as3_v4i*)lsrc, 0, 0);
#else
  unsigned loff = (unsigned)(uintptr_t)(as3_char*)lsrc;
  asm volatile("global_store_async_from_lds_b128 %0, %1, off"
               :: "v"(gdst), "v"(loff)
               : "memory");
#endif
}

__device__ __forceinline__ void wait_asynccnt0() {
#if __has_builtin(__builtin_amdgcn_s_wait_asynccnt)
  __builtin_amdgcn_s_wait_asynccnt(0);
#else
  asm volatile("s_wait_asynccnt 0" ::: "memory");
#endif
}

#define TPB    256
#define UNROLL 14   // 256 threads * 14 float4 = 3584 float4 = half of one (b,h) output row

// Input : x   (32, 224, 224, 64)  fp32
// Output: out (32, 112, 112, 256) fp32 = concat of 4 corner crops along C.
// Pure permutation/copy: 822 MB HBM traffic -> ~35us floor at 23.3 TB/s.
//
// Grid mapping (no integer division anywhere):
//   blockIdx.z = b (32), blockIdx.y = h (112), blockIdx.x = half-row (2).
//   Within a row, output float4 index widx4 = w*64 + q*16 + c4 is linear, so
//   c4 = widx4&15, q = (widx4>>4)&3, w = widx4>>6 are shifts/masks only.
__global__ __launch_bounds__(TPB)
void Sector4Pooling2D_kernel(const float* __restrict__ x, float* __restrict__ out) {
    __shared__ __align__(16) float lds[TPB * 4 * UNROLL];  // 56 KB staging

    const unsigned t    = threadIdx.x;
    const unsigned h    = blockIdx.y;
    const unsigned b    = blockIdx.z;
    const unsigned base = blockIdx.x * (TPB * UNROLL) + t;       // widx4 of this thread, u=0
    const unsigned rowbase_out4 = (b * 112u + h) * 7168u;        // row start in float4 units

    // Phase 1: gather the four input corners of this row into LDS via async B128 DMA.
#pragma unroll
    for (int u = 0; u < UNROLL; ++u) {
        const unsigned widx4 = base + (unsigned)u * TPB;          // 0..7167 within row
        const unsigned c4    = widx4 & 15u;                       // float4 within 64-ch block
        const unsigned q     = (widx4 >> 4) & 3u;                 // which corner
        const unsigned w     = widx4 >> 6;                        // 0..111
        const unsigned hh    = h + (q >> 1) * 112u;               // row in 224
        const unsigned ww    = w + (q & 1u) * 112u;               // col in 224
        const unsigned in4   = ((b * 224u + hh) * 224u + ww) * 16u + c4;
        async_g2l_b128(x + (size_t)in4 * 4u, &lds[((unsigned)u * TPB + t) * 4u]);
    }

    // Each lane only consumes LDS bytes it loaded itself: wait ASYNCcnt, no barrier.
    wait_asynccnt0();

    // Phase 2: stream LDS to the fully-contiguous 56 KB output span via async B128 DMA.
#pragma unroll
    for (int u = 0; u < UNROLL; ++u) {
        const unsigned out4 = rowbase_out4 + base + (unsigned)u * TPB;
        async_l2g_b128(out + (size_t)out4 * 4u, &lds[((unsigned)u * TPB + t) * 4u]);
    }
    // S_ENDPGM performs an implicit wait-idle, covering outstanding async stores.
}

extern "C" void kernel_launch(void* const* d_in, const int* in_sizes, int n_in,
                              void* d_out, int out_size, void* d_ws, size_t ws_size,
                              hipStream_t stream) {
    (void)in_sizes; (void)n_in; (void)d_ws; (void)ws_size; (void)out_size;
    const float* x   = (const float*)d_in[0];
    float*       out = (float*)d_out;

    // One block per half output row: 2 * 112 * 32 = 7168 blocks, each moving
    // 56 KB in + 56 KB out. Total = 32*112*112*256 floats, covered exactly.
    Sector4Pooling2D_kernel<<<dim3(2, 112, 32), dim3(TPB), 0, stream>>>(x, out);
}